// OrthogonalLinear_5695126634722
// MI455X (gfx1250) — compile-verified
//
#include <hip/hip_runtime.h>

// ---------------------------------------------------------------------------
// OrthogonalLinear on MI455X (gfx1250, wave32):
//   out = x * M + bias, where M = product of 130,816 Givens rotations.
// Phase 0: cos/sin of every theta -> ws
// Phase 1: build M by running the 1021-layer brickwork schedule on identity
//          rows (16 rows per WG, LDS-resident, bank-conflict-free stride)
// Phase 2: fp32 WMMA GEMM (V_WMMA_F32_16X16X4_F32), one wave per 16x16 tile
// ---------------------------------------------------------------------------

typedef __attribute__((ext_vector_type(2))) float v2f;
typedef __attribute__((ext_vector_type(8))) float v8f;

#define NWIRES   512          // n == m == 512 for this problem shape
#define NSTATE   514          // n + 2 dummy wires
#define SSTRIDE  516          // LDS row stride in floats; 516 % 64 == 4 -> no bank conflicts
#define NLAYERS  1021         // t = 0 .. 2*(n-2) ; last gate at t = 2*510 - 0 = 1020
#define ROWS_PER_WG 16

// ---------------- Phase 0: theta -> (cos, sin) -------------------------------
__global__ void ortho_sincos_kernel(const float* __restrict__ th,
                                    float2* __restrict__ cs, int n) {
  int i = blockIdx.x * blockDim.x + threadIdx.x;
  if (i < n) {
    float s, c;
    __sincosf(th[i], &s, &c);
    cs[i] = make_float2(c, s);
  }
}

// ---------------- Phase 1: build M rows through the rotation circuit ---------
// Schedule (closed form, n == m == 512):
//   layer t:  k in [klo, khi],  klo = (t>>1) + 1 + (t&1),  khi = min(t+1, 511)
//   gate for k:  wires (i, i+1) with i = 2*(k-1) - t
//   theta index: ti = k*(k-1)/2 + (t - k + 1)
// Each gate rotates a disjoint wire pair -> one barrier per layer suffices.
__global__ void ortho_build_kernel(const float2* __restrict__ cs,
                                   float* __restrict__ Mbuf) {
  __shared__ float S[ROWS_PER_WG][SSTRIDE];
  const int tid  = threadIdx.x;          // 256 threads = 8 waves
  const int r    = tid & 15;             // local row 0..15
  const int slot = tid >> 4;             // gate slot 0..15
  const int row0 = blockIdx.x * ROWS_PER_WG;

  // Identity rows row0..row0+15 (514 wires, dummies start at 0)
  for (int c = tid; c < NSTATE; c += 256) {
    #pragma unroll
    for (int rr = 0; rr < ROWS_PER_WG; ++rr)
      S[rr][c] = (c == row0 + rr) ? 1.0f : 0.0f;
  }
  __syncthreads();

  for (int t = 0; t < NLAYERS; ++t) {
    const int klo = (t >> 1) + 1 + (t & 1);
    const int khi = min(t + 1, NWIRES - 1);
    const int G   = khi - klo + 1;
    for (int j = slot; j < G; j += 16) {
      const int k  = klo + j;
      const int i  = 2 * (k - 1) - t;
      const int ti = (k * (k - 1)) / 2 + (t - k + 1);
      const float2 csv = cs[ti];         // broadcast across the 16 row-threads
      const float a = S[r][i];
      const float b = S[r][i + 1];
      S[r][i]     = csv.x * a + csv.y * b;
      S[r][i + 1] = csv.x * b - csv.y * a;
    }
    __syncthreads();
  }

  // Emit M rows (only the first 512 columns are ever read)
  for (int c = tid; c < NWIRES; c += 256) {
    #pragma unroll
    for (int rr = 0; rr < ROWS_PER_WG; ++rr)
      Mbuf[(size_t)(row0 + rr) * NWIRES + c] = S[rr][c];
  }
}

// ---------------- Phase 2: out = x * M + bias via fp32 WMMA ------------------
// One wave computes one 16x16 tile of out.  K-loop: 512/4 = 128 WMMA ops.
// A (16x4 f32, 2 VGPRs): lanes 0-15 hold K=0,1; lanes 16-31 hold K=2,3.
// B (4x16 f32, 2 VGPRs): mirrored striping (N across lanes).
// C/D (16x16 f32, 8 VGPRs): VGPR v -> M = v (+8 for lanes 16-31), N = lane&15.
__global__ void ortho_gemm_kernel(const float* __restrict__ X,
                                  const float* __restrict__ Mb,
                                  const float* __restrict__ bias,
                                  float* __restrict__ out,
                                  int colTiles) {
  const int wv    = threadIdx.x >> 5;                        // wave in block
  const int gw    = blockIdx.x * (blockDim.x >> 5) + wv;     // global wave id
  const int tileM = gw / colTiles;
  const int tileN = gw - tileM * colTiles;
  const int lane  = threadIdx.x & 31;
  const int half  = lane >> 4;                               // 0: K=0,1  1: K=2,3
  const int l16   = lane & 15;
  const int rowA  = tileM * 16 + l16;
  const int colB  = tileN * 16 + l16;

  const float* xrow = X + (size_t)rowA * NWIRES;

  v8f acc = {};
  #pragma unroll 4
  for (int k0 = 0; k0 < NWIRES; k0 += 4) {
    const int ka = k0 + half * 2;
    v2f a, b;
    a.x = xrow[ka];
    a.y = xrow[ka + 1];
    b.x = Mb[(size_t)ka * NWIRES + colB];
    b.y = Mb[(size_t)(ka + 1) * NWIRES + colB];
    // (neg_a, A, neg_b, B, c_mod, C, reuse_a, reuse_b)
    acc = __builtin_amdgcn_wmma_f32_16x16x4_f32(
        false, a, false, b, (short)0, acc, false, false);
  }

  const float bv = bias[colB];
  #pragma unroll
  for (int v = 0; v < 8; ++v) {
    const int m = tileM * 16 + half * 8 + v;
    out[(size_t)m * NWIRES + colB] = acc[v] + bv;
  }
}

// ---------------------------------------------------------------------------
extern "C" void kernel_launch(void* const* d_in, const int* in_sizes, int n_in,
                              void* d_out, int out_size, void* d_ws, size_t ws_size,
                              hipStream_t stream) {
  const float* x      = (const float*)d_in[0];   // (B, 512) fp32
  const float* thetas = (const float*)d_in[1];   // (130816,) fp32
  const float* bias   = (const float*)d_in[2];   // (512,) fp32
  float*       out    = (float*)d_out;           // (B, 512) fp32

  const int B   = in_sizes[0] / NWIRES;          // 256
  const int nth = in_sizes[1];                   // 130816

  // Workspace layout: [cos/sin float2 array][256B align][M 512x512 f32]
  float2* cs = (float2*)d_ws;
  size_t csBytes = ((size_t)nth * sizeof(float2) + 255) & ~(size_t)255;
  float* Mbuf = (float*)((char*)d_ws + csBytes);

  // Phase 0: sincos
  ortho_sincos_kernel<<<(nth + 255) / 256, 256, 0, stream>>>(thetas, cs, nth);

  // Phase 1: build M (32 WGs x 16 identity rows each)
  ortho_build_kernel<<<NWIRES / ROWS_PER_WG, 256, 0, stream>>>(cs, Mbuf);

  // Phase 2: WMMA GEMM, 4 waves (128 threads) per block, 1 wave per 16x16 tile
  const int colTiles = NWIRES / 16;              // 32
  const int rowTiles = B / 16;                   // 16
  const int nWaves   = rowTiles * colTiles;      // 512
  ortho_gemm_kernel<<<nWaves / 4, 128, 0, stream>>>(x, Mbuf, bias, out, colTiles);
}